// GenSP_87505663688954
// MI455X (gfx1250) — compile-verified
//
#include <hip/hip_runtime.h>
#include <hip/hip_bf16.h>
#include <stdint.h>

// ---------------------------------------------------------------------------
// GenSP superpixel affinity for MI455X (gfx1250, wave32, WMMA + TDM + async).
// B=4, C=64, H=W=256, stoken=16 -> nH=nW=16, nS=256, P=65536, N_ITER=2, M=0.
// Grid channels are zero (M_COEF=0) so effective Cf = 64.
// Memory-bound: 256 MiB dense output write dominates (~11 us at 23.3 TB/s).
// ---------------------------------------------------------------------------

#define GB   4        // batches
#define GC   64       // channels
#define GH   256
#define GW   256
#define GSH  16       // stoken
#define GNH  16
#define GNW  16
#define GNS  256      // superpixels
#define GP   (GH * GW)

typedef __attribute__((ext_vector_type(2))) float    v2f;
typedef __attribute__((ext_vector_type(4))) float    v4fn;
typedef __attribute__((ext_vector_type(8))) float    v8f;
typedef __attribute__((ext_vector_type(4))) uint32_t v4u;
typedef __attribute__((ext_vector_type(4))) int      v4i;
typedef __attribute__((ext_vector_type(8))) int      v8i;

// LDS layout (floats), dynamic shared memory:
//   sFeat : 64 * 257 = 16448  channel-major (TDM writes [ch][p], 1-dword pad
//                             per 256 dwords -> stride 257, banks (ch+p)%64)
//   sCent : 64 * 16  = 1024   channel-major, 9 cands padded to 16
//   sC2   : 16
//   sF2   : 256
//   sAff  : 16 * 260 = 4160   cand-major, stride 260 (16B-aligned rows for
//                             async b128 stores; banks (4n+p)%64)
#define OFF_FEAT 0
#define OFF_CENT 16448
#define OFF_C2   17472
#define OFF_F2   17488
#define OFF_AFF  17744
#define SMEM_FLOATS (OFF_AFF + 16 * 260)
#define SMEM_BYTES  (SMEM_FLOATS * 4)

__device__ __forceinline__ void cand_info(int bi, int bj, int n,
                                          int& cs, bool& valid) {
    int di = n / 3 - 1, dj = n % 3 - 1;
    int ci = bi + di, cj = bj + dj;
    valid = (n < 9) && (ci >= 0) && (ci < GNH) && (cj >= 0) && (cj < GNW);
    int cci = min(GNH - 1, max(0, ci));
    int ccj = min(GNW - 1, max(0, cj));
    cs = cci * GNW + ccj;
}

// ---------------------------------------------------------------------------
__global__ void zero_f32(float* __restrict__ p, size_t n) {
    size_t i = (size_t)blockIdx.x * blockDim.x + threadIdx.x;
    size_t stride = (size_t)gridDim.x * blockDim.x;
    v4fn* p4 = (v4fn*)p;
    size_t n4 = n >> 2;
    v4fn z = {0.0f, 0.0f, 0.0f, 0.0f};
    for (; i < n4; i += stride)
        __builtin_nontemporal_store(z, p4 + i);   // TH=NT: don't thrash L2
}

// ---------------------------------------------------------------------------
// Initial centroids: 16x16 block mean. grid = B*nS blocks, 64 threads (1 ch ea).
__global__ void cent_init(const float* __restrict__ x, float* __restrict__ cent) {
    int blk = blockIdx.x;            // b*256 + s
    int b = blk >> 8, s = blk & 255;
    int bi = s >> 4, bj = s & 15;
    int c = threadIdx.x;             // channel
    const float* base = x + ((size_t)b * GC + c) * (size_t)(GH * GW)
                          + (size_t)(bi * GSH) * GW + (bj * GSH);
    float acc = 0.0f;
    for (int r = 0; r < GSH; ++r) {
        const v4fn* row = (const v4fn*)(base + (size_t)r * GW);
        v4fn a = row[0], bb = row[1], cc = row[2], dd = row[3];
        acc += a.x + a.y + a.z + a.w + bb.x + bb.y + bb.z + bb.w
             + cc.x + cc.y + cc.z + cc.w + dd.x + dd.y + dd.z + dd.w;
    }
    cent[((size_t)b * GC + c) * GNS + s] = acc * (1.0f / (GSH * GSH));
}

// ---------------------------------------------------------------------------
// One affinity iteration over a (b, block) tile.
// PHASE 0: write per-block centroid partials pnum (b,blk,9,64) / pden (b,blk,9).
// PHASE 1: async-store affinities into dense output A (B, nS, P) (pre-zeroed).
template <int PHASE>
__global__ __launch_bounds__(256)
void gensp_iter(const float* __restrict__ x, const float* __restrict__ cent,
                float* __restrict__ pnum, float* __restrict__ pden,
                float* __restrict__ out) {
    extern __shared__ float smem[];
    float* sFeat = smem + OFF_FEAT;   // [ch][p], stride 257
    float* sCent = smem + OFF_CENT;   // [ch][n16]
    float* sC2   = smem + OFF_C2;
    float* sF2   = smem + OFF_F2;
    float* sAff  = smem + OFF_AFF;    // [n16][p], stride 260

    const int tid = threadIdx.x;
    const int blk = blockIdx.x;          // b*256 + s
    const int b = blk >> 8, s = blk & 255;
    const int bi = s >> 4, bj = s & 15;
    const int wave = tid >> 5, lane = tid & 31;

    // ---- TDM: DMA the 16x16x64 feature tile into LDS (wave 0 issues) ------
    if (wave == 0) {
        const uint64_t gaddr = (uint64_t)(const void*)
            (x + (size_t)b * GC * (size_t)(GH * GW)
               + (size_t)(bi * GSH) * GW + (bj * GSH));
        const uint32_t laddr = (uint32_t)(size_t)(void*)sFeat;
        v4u g0; v8i g1; v4i g2; v4i g3; v8i g4;
        // group0: count=1 | lds_addr | global_addr(57b) | type=2
        g0[0] = 1u;
        g0[1] = laddr;
        g0[2] = (uint32_t)gaddr;
        g0[3] = (uint32_t)((gaddr >> 32) & 0x01FFFFFFu) | 0x80000000u;
        // group1: data_size=4B, pad_enable, pad_interval=256dw, pad_amount=1dw
        g1[0] = (2 << 16) | (1 << 20) | (7 << 22);
        g1[1] = (int)(256u << 16);               // tensor_dim0 = 256 (lo16)
        g1[2] = (int)(256u << 16);               // td0 hi=0 | tensor_dim1=256 lo16
        g1[3] = (int)(16u << 16);                // td1 hi=0 | tile_dim0 = 16
        g1[4] = 16 | (64 << 16);                 // tile_dim1=16, tile_dim2=64
        g1[5] = GW;                              // dim0 stride = W (row)
        g1[6] = 0;                               // dim0 hi | dim1 stride lo16
        g1[7] = (GH * GW) >> 16;                 // dim1 stride = H*W (channel)
        // group2: tensor_dim2=64 (z extent), rest unused
        g2[0] = GC; g2[1] = 0; g2[2] = 0; g2[3] = 0;
        g3[0] = 0;  g3[1] = 0; g3[2] = 0; g3[3] = 0;
        g4[0] = 0;  g4[1] = 0; g4[2] = 0; g4[3] = 0;
        g4[4] = 0;  g4[5] = 0; g4[6] = 0; g4[7] = 0;
        __builtin_amdgcn_tensor_load_to_lds(g0, g1, g2, g3, g4, 0);
    }

    // ---- stage candidate centroids (9 padded to 16) while DMA runs --------
    for (int idx = tid; idx < GC * 16; idx += 256) {
        int ch = idx >> 4, n = idx & 15;
        float v = 0.0f;
        if (n < 9) {
            int cs; bool vd; cand_info(bi, bj, n, cs, vd);
            v = cent[((size_t)b * GC + ch) * GNS + cs];
        }
        sCent[ch * 16 + n] = v;
    }

    if (wave == 0) __builtin_amdgcn_s_wait_tensorcnt((short)0);
    __syncthreads();

    // ---- c2 / f2 norms ----------------------------------------------------
    if (tid < 16) {
        float acc = 0.0f;
        for (int ch = 0; ch < GC; ++ch) {
            float v = sCent[ch * 16 + tid];
            acc += v * v;
        }
        sC2[tid] = acc;
    }
    {
        float acc = 0.0f;
        #pragma unroll 8
        for (int ch = 0; ch < GC; ++ch) {
            float v = sFeat[ch * 257 + tid];
            acc += v * v;
        }
        sF2[tid] = acc;
    }
    __syncthreads();

    // ---- WMMA dot: (16 px x 64 ch) x (64 ch x 16 cand), fp32 K=4 steps ----
    const int n = lane & 15;
    const int kb = (lane >> 4) << 1;          // K offset 0 or 2 per lane-half
    int cs_n; bool valid_n; cand_info(bi, bj, n, cs_n, valid_n);
    const float c2n = sC2[n];

    for (int half = 0; half < 2; ++half) {
        const int m0 = wave * 32 + half * 16;
        const int mm = m0 + (lane & 15);
        v8f acc = {};
        #pragma unroll
        for (int k0 = 0; k0 < GC; k0 += 4) {
            v2f a, bm;
            a.x  = sFeat[(k0 + kb) * 257 + mm];
            a.y  = sFeat[(k0 + kb + 1) * 257 + mm];
            bm.x = sCent[(k0 + kb) * 16 + n];
            bm.y = sCent[(k0 + kb + 1) * 16 + n];
            acc = __builtin_amdgcn_wmma_f32_16x16x4_f32(
                false, a, false, bm, (short)0, acc, false, false);
        }
        // softmax over the 9 candidates (N lives across 16 lanes)
        #pragma unroll
        for (int r = 0; r < 8; ++r) {
            int m = m0 + r + ((lane < 16) ? 0 : 8);
            float d = acc[r];
            float logit = valid_n ? -(sF2[m] + c2n - 2.0f * d) : -1e30f;
            float mx = logit;
            #pragma unroll
            for (int dm = 1; dm < 16; dm <<= 1)
                mx = fmaxf(mx, __shfl_xor(mx, dm, 32));
            float e = valid_n ? __expf(logit - mx) : 0.0f;
            float se = e;
            #pragma unroll
            for (int dm = 1; dm < 16; dm <<= 1)
                se += __shfl_xor(se, dm, 32);
            sAff[n * 260 + m] = e / se;
        }
    }
    __syncthreads();

    if (PHASE == 0) {
        // centroid-update partials: (64 ch x 256 px) x (256 px x 16 cand)
        if (wave < 4) {
            const int cc = wave * 16 + (lane & 15);   // channel (M dim)
            v8f acc = {};
            for (int p0 = 0; p0 < 256; p0 += 4) {
                int p = p0 + kb;
                v2f a, bm;
                a.x  = sFeat[cc * 257 + p];
                a.y  = sFeat[cc * 257 + p + 1];
                bm.x = sAff[n * 260 + p];
                bm.y = sAff[n * 260 + p + 1];
                acc = __builtin_amdgcn_wmma_f32_16x16x4_f32(
                    false, a, false, bm, (short)0, acc, false, false);
            }
            if (valid_n) {
                #pragma unroll
                for (int r = 0; r < 8; ++r) {
                    int c = wave * 16 + r + ((lane < 16) ? 0 : 8);
                    pnum[(((size_t)b * GNS + s) * 9 + n) * GC + c] = acc[r];
                }
            }
        } else if (wave == 4 && lane < 9) {
            float sd = 0.0f;
            for (int p = 0; p < 256; ++p) sd += sAff[lane * 260 + p];
            pden[((size_t)b * GNS + s) * 9 + lane] = sd;
        }
    } else {
        // async LDS->global: 9 candidate rows x 64 chunks of 16B each.
        for (int it = tid; it < 9 * 64; it += 256) {
            int k  = it >> 6;
            int p4 = (it & 63) << 2;             // first pixel of 4-px chunk
            int csk; bool vk; cand_info(bi, bj, k, csk, vk);
            if (!vk) continue;
            size_t gp = (size_t)(bi * GSH + (p4 >> 4)) * GW
                      + (size_t)(bj * GSH + (p4 & 15));
            uint64_t gaddr = (uint64_t)(void*)
                (out + ((size_t)b * GNS + csk) * (size_t)GP + gp);
            uint32_t laddr = (uint32_t)(size_t)(void*)&sAff[k * 260 + p4];
            asm volatile("global_store_async_from_lds_b128 %0, %1, off"
                         :: "v"(gaddr), "v"(laddr) : "memory");
        }
        asm volatile("s_wait_asynccnt 0x0" ::: "memory");
    }
}

// ---------------------------------------------------------------------------
// Deterministic centroid gather: cent[b,c,s] = sum of the 9 neighbor-block
// partials / (den + 1e-16). Inverse of the 3x3 candidate map.
__global__ void cent_update(const float* __restrict__ pnum,
                            const float* __restrict__ pden,
                            float* __restrict__ cent) {
    int idx = blockIdx.x * 256 + threadIdx.x;   // (b, s, c), c fastest
    int c = idx & 63;
    int rest = idx >> 6;
    int s = rest & 255;
    int b = rest >> 8;
    int si = s >> 4, sj = s & 15;
    float num = 0.0f, den = 0.0f;
    #pragma unroll
    for (int k = 0; k < 9; ++k) {
        int di = k / 3 - 1, dj = k % 3 - 1;
        int oi = si - di, oj = sj - dj;
        if (oi < 0 || oi >= GNH || oj < 0 || oj >= GNW) continue;
        int src = oi * GNW + oj;
        num += pnum[(((size_t)b * GNS + src) * 9 + k) * GC + c];
        den += pden[((size_t)b * GNS + src) * 9 + k];
    }
    cent[((size_t)b * GC + c) * GNS + s] = num / (den + 1e-16f);
}

// ---------------------------------------------------------------------------
extern "C" void kernel_launch(void* const* d_in, const int* in_sizes, int n_in,
                              void* d_out, int out_size, void* d_ws, size_t ws_size,
                              hipStream_t stream) {
    const float* x = (const float*)d_in[0];
    float* out = (float*)d_out;

    // workspace layout (floats)
    float* ws   = (float*)d_ws;
    float* cent = ws;                         // B*64*256   = 65536
    float* pnum = ws + 65536;                 // B*256*9*64 = 589824
    float* pden = ws + 65536 + 589824;        // B*256*9    = 9216

    // 1) zero the dense output (256 MiB), non-temporal
    zero_f32<<<4096, 256, 0, stream>>>(out, (size_t)GB * GNS * GP);

    // 2) initial centroids = block means
    cent_init<<<GB * GNS, GC, 0, stream>>>(x, cent);

    // 3) iteration 0: affinity + centroid partials
    gensp_iter<0><<<GB * GNS, 256, SMEM_BYTES, stream>>>(x, cent, pnum, pden, nullptr);

    // 4) deterministic centroid update
    cent_update<<<(GB * GNS * GC) / 256, 256, 0, stream>>>(pnum, pden, cent);

    // 5) iteration 1: affinity -> dense A output (async LDS->global)
    gensp_iter<1><<<GB * GNS, 256, SMEM_BYTES, stream>>>(x, cent, nullptr, nullptr, out);
}